// HistoricalPromptDecoder_39324720562831
// MI455X (gfx1250) — compile-verified
//
#include <hip/hip_runtime.h>
#include <hip/hip_bf16.h>
#include <math.h>

typedef __attribute__((ext_vector_type(2))) float v2f;
typedef __attribute__((ext_vector_type(8))) float v8f;

#define CK      64
#define CV      512
#define THW_    57600
#define NPB     576          // queries (H*W) per batch
#define BATCH   2
#define TOPK    20
#define OC      1024
#define NEG_INF (-3.402823466e38f)
#define TCHUNK  128          // t rows staged per iteration (8 waves x 16)

// ---- CDNA5 async global->LDS copy (ASYNCcnt) with portable fallbacks -------
#if defined(__has_builtin)
#  if __has_builtin(__builtin_amdgcn_global_load_async_to_lds_b128)
#    define HAVE_ASYNC_LDS 1
#  endif
#  if __has_builtin(__builtin_amdgcn_s_wait_asynccnt)
#    define HAVE_WAIT_ASYNC 1
#  endif
#endif

typedef int v4i_ __attribute__((vector_size(4 * sizeof(int))));
typedef __attribute__((address_space(1))) v4i_* glob_b128_t;
typedef __attribute__((address_space(3))) v4i_* lds_b128_t;

__device__ __forceinline__ void async_copy16(const float* gp, float* lp) {
#if defined(HAVE_ASYNC_LDS)
  __builtin_amdgcn_global_load_async_to_lds_b128(
      (glob_b128_t)const_cast<float*>(gp), (lds_b128_t)lp, 0, 0);
#else
  *(float4*)lp = *(const float4*)gp;   // synchronous fallback
#endif
}

__device__ __forceinline__ void wait_async_le8() {
#if defined(HAVE_WAIT_ASYNC)
  __builtin_amdgcn_s_wait_asynccnt(8);
#elif defined(HAVE_ASYNC_LDS)
  asm volatile("s_wait_asynccnt 0x8" ::: "memory");
#endif
}

__device__ __forceinline__ void wait_async_le0() {
#if defined(HAVE_WAIT_ASYNC)
  __builtin_amdgcn_s_wait_asynccnt(0);
#elif defined(HAVE_ASYNC_LDS)
  asm volatile("s_wait_asynccnt 0x0" ::: "memory");
#endif
}

// ---------------------------------------------------------------------------
// Kernel A: affinity GEMM (v_wmma_f32_16x16x4_f32) fused with top-k + softmax.
// grid = BATCH * (NPB/16) = 72 blocks, 256 threads (8 waves).
// mk is streamed through a double-buffered LDS tile [64c x 128t] filled with
// GLOBAL_LOAD_ASYNC_TO_LDS_B128 (8 x b128 per thread per chunk), paced by
// s_wait_asynccnt; the WMMA A-fragments are read back via ds_load.
// WMMA f32 16x16x4 layouts (wave32):
//   A(m,k):  lane = m + 16*(k/2),  vgpr = k%2
//   B(k,n):  lane = n + 16*(k/2),  vgpr = k%2
//   C(m,n):  vgpr r, lanes 0-15 -> m=r, lanes 16-31 -> m=r+8; n = lane%16
// ---------------------------------------------------------------------------
__global__ __launch_bounds__(256) void aff_topk_kernel(
    const float* __restrict__ qk,   // [B, CK, NPB]
    const float* __restrict__ mk,   // [B, CK, THW]
    float* __restrict__ topw,       // [B*NPB, TOPK]
    int*   __restrict__ topi)       // [B*NPB, TOPK]
{
  // 64 KB LDS: double-buffered mk tile; merge arrays reuse it after the loop.
  __shared__ __align__(16) unsigned char smem[2 * CK * TCHUNK * 4];
  float* smk = (float*)smem;                         // [buf][c][t], stride 128
  float* sv  = (float*)smem;                         // 16*16*20 f32 (merge)
  int*   si  = (int*)(smem + 16 * 16 * TOPK * 4);    // 16*16*20 i32 (merge)

  const int tid  = threadIdx.x;
  const int w    = tid >> 5;          // wave id 0..7
  const int lane = tid & 31;
  const int m    = lane & 15;         // A row / B col within tile
  const int kh   = lane >> 4;         // K half-pair selector

  const int b  = blockIdx.x / (NPB / 16);
  const int nt = blockIdx.x % (NPB / 16);
  const int n0 = nt * 16;
  const int n  = n0 + m;

  const float* mkb = mk + (size_t)b * CK * THW_;

  // Preload the qk B-tile: constant over the whole THW loop. 16 k-steps of 4.
  float bx[16], by[16];
#pragma unroll
  for (int kk = 0; kk < 16; ++kk) {
    const int c0 = kk * 4 + 2 * kh;
    bx[kk] = qk[(b * CK + c0) * NPB + n];
    by[kk] = qk[(b * CK + c0 + 1) * NPB + n];
  }

  // Stage one 64c x 128t chunk: 2048 16-byte units, 8 per thread, contiguous
  // along t (fully coalesced), identical [c][t] layout in LDS.
  auto stage = [&](int it, int buf) {
    const int tbase = it * TCHUNK;
    float* dst = smk + buf * (CK * TCHUNK);
#pragma unroll
    for (int op = 0; op < 8; ++op) {
      const int u  = tid * 8 + op;     // 0..2047
      const int c  = u >> 5;           // 0..63
      const int tu = (u & 31) << 2;    // 0,4,...,124
      async_copy16(mkb + (size_t)c * THW_ + tbase + tu,
                   dst + c * TCHUNK + tu);
    }
  };

  // Per-thread top-k (registers, fully unrolled). This thread only ever sees
  // column n = lane%16; rows t = it*128 + w*16 + r + 8*kh.
  float tv[TOPK];
  int   ti[TOPK];
#pragma unroll
  for (int j = 0; j < TOPK; ++j) { tv[j] = NEG_INF; ti[j] = 0; }
  float thr = NEG_INF;

  const int NIT = THW_ / TCHUNK;       // 450
  const int tl  = w * 16 + m;          // this lane's t row within the chunk

  stage(0, 0);
  for (int it = 0; it < NIT; ++it) {
    if (it + 1 < NIT) { stage(it + 1, (it + 1) & 1); wait_async_le8(); }
    else              { wait_async_le0(); }
    __syncthreads();                   // stage `it` visible to all waves

    const float* buf = smk + (it & 1) * (CK * TCHUNK);
    const int tbase  = it * TCHUNK + w * 16;

    v8f   acc = {};
    float asq = 0.0f;
#pragma unroll
    for (int kk = 0; kk < 16; ++kk) {
      const int c0 = kk * 4 + 2 * kh;
      const float ax = buf[c0 * TCHUNK + tl];
      const float ay = buf[(c0 + 1) * TCHUNK + tl];
      asq += ax * ax + ay * ay;
      v2f a; a[0] = ax; a[1] = ay;
      v2f bb; bb[0] = bx[kk]; bb[1] = by[kk];
      acc = __builtin_amdgcn_wmma_f32_16x16x4_f32(
          false, a, false, bb, (short)0, acc, false, false);
    }
    // combine |mk[t]|^2 partials: lanes m and m+16 each hold half the channels
    asq += __shfl_xor(asq, 16, 32);

#pragma unroll
    for (int r = 0; r < 8; ++r) {
      const int   trow  = tbase + r + 8 * kh;            // C row for vgpr r
      const float asq_r = __shfl(asq, r + 8 * kh, 32);   // row's |mk|^2
      const float val   = (2.0f * acc[r] - asq_r) * 0.125f;  // /sqrt(64)
      if (val > thr) {
        tv[TOPK - 1] = val; ti[TOPK - 1] = trow;
#pragma unroll
        for (int j = TOPK - 1; j > 0; --j) {
          if (tv[j] > tv[j - 1]) {
            const float fv = tv[j]; tv[j] = tv[j - 1]; tv[j - 1] = fv;
            const int   iv = ti[j]; ti[j] = ti[j - 1]; ti[j - 1] = iv;
          }
        }
        thr = tv[TOPK - 1];
      }
    }
    __syncthreads();                   // done reading buf before it is re-filled
  }

  // Merge: per column, 16 contributing threads (8 waves x 2 lane-halves).
  const int col  = m;
  const int slot = w * 2 + kh;
  const int base = (col * 16 + slot) * TOPK;
#pragma unroll
  for (int j = 0; j < TOPK; ++j) { sv[base + j] = tv[j]; si[base + j] = ti[j]; }
  __syncthreads();

  if (tid < 16) {
    const int c  = tid;
    const int cb = c * 16 * TOPK;
    float outv[TOPK]; int outi[TOPK];
    for (int e = 0; e < TOPK; ++e) {
      float bv = NEG_INF; int bj = 0;
      for (int j = 0; j < 16 * TOPK; ++j) {
        const float v = sv[cb + j];
        if (v > bv) { bv = v; bj = j; }
      }
      outv[e] = bv; outi[e] = si[cb + bj];
      sv[cb + bj] = NEG_INF;
    }
    // softmax over the 20 kept values (outv[0] is the max)
    float s = 0.0f;
    float ex[TOPK];
    for (int j = 0; j < TOPK; ++j) { ex[j] = __expf(outv[j] - outv[0]); s += ex[j]; }
    const float inv = 1.0f / s;
    const int pix = b * NPB + n0 + c;
    for (int j = 0; j < TOPK; ++j) {
      topw[pix * TOPK + j] = ex[j] * inv;
      topi[pix * TOPK + j] = outi[j];
    }
  }
}

// ---------------------------------------------------------------------------
// Kernel B: sparse readout (20 mv columns) + residual + LayerNorm(C).
// grid = B*NPB = 1152 blocks, 256 threads (each handles 2 channels).
// ---------------------------------------------------------------------------
__global__ __launch_bounds__(256) void readout_ln_kernel(
    const float* __restrict__ mv,     // [B, CV, THW]
    const float* __restrict__ qv,     // [B, CV, NPB]
    const float* __restrict__ gamma,  // [CV]
    const float* __restrict__ beta,   // [CV]
    const float* __restrict__ topw,
    const int*   __restrict__ topi,
    float* __restrict__ fh)           // [B*NPB, CV]
{
  const int p = blockIdx.x;
  const int b = p / NPB;
  const int n = p % NPB;
  const int tid  = threadIdx.x;
  const int w    = tid >> 5;
  const int lane = tid & 31;

  __shared__ float swgt[TOPK];
  __shared__ int   sidx[TOPK];
  __shared__ float rsum[8], rsq[8];
  if (tid < TOPK) { swgt[tid] = topw[p * TOPK + tid]; sidx[tid] = topi[p * TOPK + tid]; }
  __syncthreads();

  const int c0 = tid;         // channel 0..255
  const int c1 = tid + 256;   // channel 256..511
  const size_t mvb = (size_t)b * CV * THW_;

  float x0 = 0.0f, x1 = 0.0f;
#pragma unroll
  for (int j = 0; j < TOPK; ++j) {
    const float wj = swgt[j];
    const int   t  = sidx[j];
    x0 += wj * mv[mvb + (size_t)c0 * THW_ + t];
    x1 += wj * mv[mvb + (size_t)c1 * THW_ + t];
  }
  x0 += qv[((size_t)b * CV + c0) * NPB + n];
  x1 += qv[((size_t)b * CV + c1) * NPB + n];

  // block reduction of sum and sum-of-squares over 512 channels
  float s  = x0 + x1;
  float s2 = x0 * x0 + x1 * x1;
#pragma unroll
  for (int off = 16; off > 0; off >>= 1) {
    s  += __shfl_down(s,  off, 32);
    s2 += __shfl_down(s2, off, 32);
  }
  if (lane == 0) { rsum[w] = s; rsq[w] = s2; }
  __syncthreads();
  if (tid == 0) {
    float a = 0.0f, q = 0.0f;
#pragma unroll
    for (int i = 0; i < 8; ++i) { a += rsum[i]; q += rsq[i]; }
    rsum[0] = a; rsq[0] = q;
  }
  __syncthreads();
  const float mu  = rsum[0] * (1.0f / CV);
  const float var = rsq[0] * (1.0f / CV) - mu * mu;
  const float inv = rsqrtf(var + 1e-5f);

  fh[(size_t)p * CV + c0] = (x0 - mu) * inv * gamma[c0] + beta[c0];
  fh[(size_t)p * CV + c1] = (x1 - mu) * inv * gamma[c1] + beta[c1];
}

// ---------------------------------------------------------------------------
// Kernel C: 1x1 conv as WMMA f32 GEMM: out[p,o] = fh[p,:] . conv_w[o,:] + b[o].
// 72 pixel-tiles x 64 out-tiles = 4608 wave-tiles; 8 waves/block -> 576 blocks.
// Both A and B K-pairs are contiguous in memory -> float2 loads.
// ---------------------------------------------------------------------------
__global__ __launch_bounds__(256) void conv_wmma_kernel(
    const float* __restrict__ fh,    // [B*NPB, CV]
    const float* __restrict__ cw,    // [OC, CV]
    const float* __restrict__ cb,    // [OC]
    float* __restrict__ out)         // [B, OC, NPB]
{
  const int tid  = threadIdx.x;
  const int w    = tid >> 5;
  const int lane = tid & 31;
  const int m    = lane & 15;
  const int kh   = lane >> 4;

  const int tile = blockIdx.x * 8 + w;
  const int mt   = tile / (OC / 16);
  const int nt   = tile % (OC / 16);
  const int p0   = mt * 16;
  const int o0   = nt * 16;

  const float* arow = fh + (size_t)(p0 + m) * CV;
  const float* brow = cw + (size_t)(o0 + m) * CV;

  v8f acc = {};
#pragma unroll 4
  for (int kk = 0; kk < CV / 4; ++kk) {
    const int c0 = kk * 4 + 2 * kh;
    const float2 av = *(const float2*)&arow[c0];
    const float2 bv = *(const float2*)&brow[c0];
    v2f a; a[0] = av.x; a[1] = av.y;
    v2f bb; bb[0] = bv.x; bb[1] = bv.y;
    acc = __builtin_amdgcn_wmma_f32_16x16x4_f32(
        false, a, false, bb, (short)0, acc, false, false);
  }

  const int   o    = o0 + m;
  const float bias = cb[o];
#pragma unroll
  for (int r = 0; r < 8; ++r) {
    const int pix = p0 + r + 8 * kh;     // C row
    const int bb_ = pix / NPB;
    const int nn  = pix % NPB;
    out[((size_t)bb_ * OC + o) * NPB + nn] = acc[r] + bias;
  }
}

// ---------------------------------------------------------------------------
extern "C" void kernel_launch(void* const* d_in, const int* in_sizes, int n_in,
                              void* d_out, int out_size, void* d_ws, size_t ws_size,
                              hipStream_t stream) {
  (void)in_sizes; (void)n_in; (void)out_size; (void)ws_size;
  const float* qk    = (const float*)d_in[0];
  const float* mk    = (const float*)d_in[1];
  const float* mv    = (const float*)d_in[2];
  const float* qv    = (const float*)d_in[3];
  const float* gamma = (const float*)d_in[4];
  const float* beta  = (const float*)d_in[5];
  const float* cw    = (const float*)d_in[6];
  const float* cb    = (const float*)d_in[7];
  // d_in[8] = top_k (== 20, compile-time constant here)

  char*  ws   = (char*)d_ws;
  const size_t seg = ((size_t)BATCH * NPB * TOPK * 4 + 255) & ~(size_t)255;
  float* topw = (float*)ws;                  // 1152*20 f32
  int*   topi = (int*)(ws + seg);            // 1152*20 i32
  float* fh   = (float*)(ws + 2 * seg);      // 1152*512 f32
  float* out  = (float*)d_out;

  aff_topk_kernel<<<BATCH * (NPB / 16), 256, 0, stream>>>(qk, mk, topw, topi);
  readout_ln_kernel<<<BATCH * NPB, 256, 0, stream>>>(mv, qv, gamma, beta, topw, topi, fh);
  conv_wmma_kernel<<<(BATCH * NPB / 16) * (OC / 16) / 8, 256, 0, stream>>>(fh, cw, cb, out);
}